// LengthRegulator_63866163692335
// MI455X (gfx1250) — compile-verified
//
#include <hip/hip_runtime.h>
#include <stdint.h>

// Problem constants (fixed by the reference).
#define B        64
#define T        256
#define D        384                   // floats per row = 96 float4
#define MAX_DUR  8
#define T_OUT    (T * MAX_DUR)         // 2048
#define ROWS_PER_BLOCK 32              // == one wave32 of searches
#define ROWS_PER_WAVE  4               // 8 waves * 4 rows = 32 rows

typedef float vfloat4 __attribute__((ext_vector_type(4)));

// clang's gfx1250 async builtin takes typed AS pointers: AS1 src, AS3 dst.
typedef __attribute__((address_space(1))) int gbl_int_t;
typedef __attribute__((address_space(3))) int lds_int_t;

__device__ __forceinline__ void async_lds_load_i32(const int* gsrc, int* lds_dst) {
#if __has_builtin(__builtin_amdgcn_global_load_async_to_lds_b32)
  __builtin_amdgcn_global_load_async_to_lds_b32(
      (gbl_int_t*)(uintptr_t)gsrc,
      (lds_int_t*)(uint32_t)(uintptr_t)lds_dst,
      /*offset=*/0, /*cpol=*/0);
#else
  uint32_t lds_addr = (uint32_t)(uintptr_t)lds_dst;
  asm volatile("global_load_async_to_lds_b32 %0, %1, off"
               :: "v"(lds_addr), "v"(gsrc) : "memory");
#endif
#if __has_builtin(__builtin_amdgcn_s_wait_asynccnt)
  __builtin_amdgcn_s_wait_asynccnt(0);
#else
  asm volatile("s_wait_asynccnt 0x0" ::: "memory");
#endif
}

// ---------------- Kernel A: ends[b,:] = cumsum(ds[b,:]) ----------------
__global__ __launch_bounds__(256) void scan_kernel(
    const int* __restrict__ ds, int* __restrict__ ends)
{
  __shared__ int s[T];
  const int tid = threadIdx.x;
  const int b   = blockIdx.x;

  async_lds_load_i32(ds + (size_t)b * T + tid, &s[tid]);
  __syncthreads();

  #pragma unroll
  for (int off = 1; off < T; off <<= 1) {
    int add = (tid >= off) ? s[tid - off] : 0;
    __syncthreads();
    s[tid] += add;
    __syncthreads();
  }
  ends[(size_t)b * T + tid] = s[tid];
}

// ---------------- Kernel B: dense gather ----------------
// PRE=true : `tbl` holds pre-scanned ends (from d_ws) -> no in-block scan.
// PRE=false: `tbl` holds raw ds -> fused scan fallback (if ws too small).
template <bool PRE>
__global__ __launch_bounds__(256) void gather_kernel(
    const float* __restrict__ xs,   // (B, T, D)
    const int*   __restrict__ tbl,  // (B, T) ends (PRE) or ds (!PRE)
    float*       __restrict__ out)  // (B, T_OUT, D)
{
  __shared__ int s_ends[T];
  __shared__ int s_idx[ROWS_PER_BLOCK];   // source row, or -1 for PAD

  const int tid = threadIdx.x;
  const int b   = blockIdx.y;

  async_lds_load_i32(tbl + (size_t)b * T + tid, &s_ends[tid]);
  __syncthreads();

  if (!PRE) {   // fused inclusive scan fallback
    #pragma unroll
    for (int off = 1; off < T; off <<= 1) {
      int add = (tid >= off) ? s_ends[tid - off] : 0;
      __syncthreads();
      s_ends[tid] += add;
      __syncthreads();
    }
  }

  // Wave 0: each lane binary-searches one of this block's 32 output rows.
  if (tid < ROWS_PER_BLOCK) {
    const int t = blockIdx.x * ROWS_PER_BLOCK + tid;
    const int total = s_ends[T - 1];
    // searchsorted(ends, t, side='right'): count of ends[j] <= t.
    int n = 0;
    #pragma unroll
    for (int s = 128; s >= 1; s >>= 1) {
      int c = n + s;                       // c in [1, 255]
      if (s_ends[c - 1] <= t) n = c;
    }
    if (n > T - 1) n = T - 1;              // clamp (matches reference)
    s_idx[tid] = (t < total) ? n : -1;
  }
  __syncthreads();

  // 8 waves x 4 rows: stream rows with b128 loads + non-temporal b128 stores.
  const int wave = tid >> 5;               // wave32
  const int lane = tid & 31;
  const int row0 = blockIdx.x * ROWS_PER_BLOCK + wave * ROWS_PER_WAVE;

  #pragma unroll
  for (int i = 0; i < ROWS_PER_WAVE; ++i) {
    const int t = row0 + i;
    const int n = s_idx[wave * ROWS_PER_WAVE + i];
    vfloat4* dst = (vfloat4*)(out + ((size_t)b * T_OUT + t) * D);

    if (n >= 0) {
      const vfloat4* src = (const vfloat4*)(xs + ((size_t)b * T + n) * D);
      #pragma unroll
      for (int k = 0; k < 3; ++k) {        // 3 x 32 lanes x 16B = 1536B row
        vfloat4 v = src[lane + 32 * k];    // global_load_b128 (L2-resident xs)
        __builtin_nontemporal_store(v, dst + lane + 32 * k);  // NT b128 store
      }
    } else {
      vfloat4 z = {0.f, 0.f, 0.f, 0.f};    // PAD_VALUE
      #pragma unroll
      for (int k = 0; k < 3; ++k)
        __builtin_nontemporal_store(z, dst + lane + 32 * k);
    }
  }
}

extern "C" void kernel_launch(void* const* d_in, const int* in_sizes, int n_in,
                              void* d_out, int out_size, void* d_ws, size_t ws_size,
                              hipStream_t stream) {
  const float* xs  = (const float*)d_in[0];
  const int*   ds  = (const int*)d_in[1];
  float*       out = (float*)d_out;

  dim3 grid(T_OUT / ROWS_PER_BLOCK, B);    // (64, 64)

  const size_t ends_bytes = (size_t)B * T * sizeof(int);   // 64 KB
  if (d_ws != nullptr && ws_size >= ends_bytes) {
    int* ends = (int*)d_ws;
    scan_kernel<<<B, 256, 0, stream>>>(ds, ends);
    gather_kernel<true><<<grid, 256, 0, stream>>>(xs, ends, out);
  } else {
    gather_kernel<false><<<grid, 256, 0, stream>>>(xs, ds, out);
  }
}